// LocalAttention2d_19327352832727
// MI455X (gfx1250) — compile-verified
//
#include <hip/hip_runtime.h>
#include <hip/hip_bf16.h>
#include <math.h>

typedef __attribute__((ext_vector_type(2))) float v2f;
typedef __attribute__((ext_vector_type(8))) float v8f;

#define B_    8
#define H_    64
#define W_    64
#define D_    256
#define N_    1024
#define ROWS_ 65
#define COLS_ 65
#define KW_   15          // (2*1+1)*(2*2+1)

// -----------------------------------------------------------------------------
// Kernel 1: V[m,n] = sum_k A[m,k] * Wm[k,n]
//   A  = c_t  [M=8192, K=256] row-major
//   Wm = W_a  [K=256, N=256]  row-major   (v = W_a^T c  ->  B[k,n] = W_a[k,n])
// One wave per 16x64 C-tile: 4 f32 WMMA accumulators share one A fragment,
// quadrupling A reuse and WMMA density per issued load.
// -----------------------------------------------------------------------------
__global__ __launch_bounds__(256) void gemm_f32_wmma(const float* __restrict__ A,
                                                     const float* __restrict__ Wm,
                                                     float* __restrict__ V)
{
    const int lane  = threadIdx.x & 31;
    const int wave  = threadIdx.x >> 5;
    const int tile  = blockIdx.x * 8 + wave;      // 2048 wave-tiles total
    const int tm    = tile >> 2;                  // 512 row tiles (16 rows each)
    const int n0    = (tile & 3) * 64;            // 4 column stripes of 64
    const int m0    = tm * 16;
    const int row   = lane & 15;
    const int khalf = (lane >> 4) << 1;           // lanes 0-15: K=0,1 ; lanes 16-31: K=2,3

    const float* Arow = A  + (size_t)(m0 + row) * D_;
    const float* Bcol = Wm + (size_t)(n0 + row);  // +j*16 selects the 4 sub-tiles

    v8f acc0 = {}, acc1 = {}, acc2 = {}, acc3 = {};
#pragma unroll 2
    for (int k = 0; k < D_; k += 4) {
        const v2f a = *(const v2f*)(Arow + k + khalf);        // b64 load, contiguous K pair
        const size_t r0 = (size_t)(k + khalf)     * D_;
        const size_t r1 = (size_t)(k + khalf + 1) * D_;

        v2f b0; b0.x = Bcol[r0];      b0.y = Bcol[r1];
        v2f b1; b1.x = Bcol[r0 + 16]; b1.y = Bcol[r1 + 16];
        v2f b2; b2.x = Bcol[r0 + 32]; b2.y = Bcol[r1 + 32];
        v2f b3; b3.x = Bcol[r0 + 48]; b3.y = Bcol[r1 + 48];

        acc0 = __builtin_amdgcn_wmma_f32_16x16x4_f32(false, a, false, b0, (short)0, acc0, false, false);
        acc1 = __builtin_amdgcn_wmma_f32_16x16x4_f32(false, a, false, b1, (short)0, acc1, false, false);
        acc2 = __builtin_amdgcn_wmma_f32_16x16x4_f32(false, a, false, b2, (short)0, acc2, false, false);
        acc3 = __builtin_amdgcn_wmma_f32_16x16x4_f32(false, a, false, b3, (short)0, acc3, false, false);
    }

    // C/D layout: VGPR i -> M = i (+8 for lanes 16-31), N = lane & 15
    const int mtop = (lane >> 4) * 8;
    float* out = V + (size_t)(m0 + mtop) * D_ + n0 + row;
#pragma unroll
    for (int i = 0; i < 8; ++i) {
        out[(size_t)i * D_]      = acc0[i];
        out[(size_t)i * D_ + 16] = acc1[i];
        out[(size_t)i * D_ + 32] = acc2[i];
        out[(size_t)i * D_ + 48] = acc3[i];
    }
}

// -----------------------------------------------------------------------------
// Kernel 2: per (b,n) local window attention. One wave32 per point.
//   qg_k gathered from q with NaN-pad semantics (pad row0/col0 -> invalid),
//   a_k = qg_k . v,   softmax (invalid -> -inf) * gaussian,  out = sum w_k qg_k.
// Window vectors cached in LDS between the score pass and the weighted-sum pass.
// -----------------------------------------------------------------------------
__global__ __launch_bounds__(128) void local_attn(const float* __restrict__ q,
                                                  const float* __restrict__ p_t,
                                                  const float* __restrict__ v,
                                                  float* __restrict__ out)
{
    const int lane  = threadIdx.x & 31;
    const int wave  = threadIdx.x >> 5;
    const int point = blockIdx.x * 4 + wave;      // b*N + n, 8192 total
    const int b     = point >> 10;                // / N_

    extern __shared__ float lds[];
    float* myq = lds + (size_t)wave * (KW_ * D_);

    const float p0 = p_t[point * 2 + 0];
    const float p1 = p_t[point * 2 + 1];
    const int p0i = (int)p0;
    const int p1i = (int)p1;

    // per-lane chunk of v[b,n,:]: floats [4*lane..4*lane+3] and [128+4*lane..]
    const float4* vrow = (const float4*)(v + (size_t)point * D_);
    const float4 vlo = vrow[lane];
    const float4 vhi = vrow[lane + 32];

    // window indices + gaussian factors (replicated across lanes)
    int   rr[3], cc[5];
    float rexp[3], cexp[5];
#pragma unroll
    for (int i = 0; i < 3; ++i) {
        int r = p0i + i;                          // p0i + (i-1) + 1
        r = r < 0 ? 0 : (r > ROWS_ ? ROWS_ : r);
        r = r % ROWS_;
        rr[i] = r;
        float dr = (float)((r - 1) > 0 ? (r - 1) : 0) - p0;   // R_WIN = 1
        rexp[i] = __expf(-2.0f * dr * dr);
    }
#pragma unroll
    for (int j = 0; j < 5; ++j) {
        int c = p1i + j - 1;                      // p1i + (j-2) + 1
        c = c < 0 ? 0 : (c > COLS_ ? COLS_ : c);
        c = c % COLS_;
        cc[j] = c;
        float dc = ((float)((c - 1) > 0 ? (c - 1) : 0) - p1) * 0.5f;  // C_WIN = 2
        cexp[j] = __expf(-2.0f * dc * dc);
    }

    const float* qb = q + (size_t)b * (H_ * W_ * D_);

    // Pass 1: gather windows to LDS + score dots
    float a[KW_];
#pragma unroll
    for (int k = 0; k < KW_; ++k) {
        const int i = k / 5, j = k % 5;
        const int r = rr[i], c = cc[j];
        const bool valid = (r > 0) && (c > 0);    // pad row/col are the NaN slots
        float4 lo = make_float4(0.f, 0.f, 0.f, 0.f);
        float4 hi = make_float4(0.f, 0.f, 0.f, 0.f);
        if (valid) {
            const float4* src =
                (const float4*)(qb + ((size_t)(r - 1) * W_ + (size_t)(c - 1)) * D_);
            lo = src[lane];
            hi = src[lane + 32];
        }
        float4* dst = (float4*)(myq + k * D_);
        dst[lane]      = lo;
        dst[lane + 32] = hi;

        float d = lo.x * vlo.x + lo.y * vlo.y + lo.z * vlo.z + lo.w * vlo.w
                + hi.x * vhi.x + hi.y * vhi.y + hi.z * vhi.z + hi.w * vhi.w;
#pragma unroll
        for (int off = 16; off > 0; off >>= 1)
            d += __shfl_xor(d, off, 32);
        a[k] = valid ? d : -INFINITY;
    }

    // softmax over the 15 scores (replicated on all 32 lanes) * gaussian decay
    float m = a[0];
#pragma unroll
    for (int k = 1; k < KW_; ++k) m = fmaxf(m, a[k]);
    float e[KW_];
    float s = 0.f;
#pragma unroll
    for (int k = 0; k < KW_; ++k) { e[k] = __expf(a[k] - m); s += e[k]; }
    const float inv = 1.0f / s;
    float w[KW_];
#pragma unroll
    for (int k = 0; k < KW_; ++k)
        w[k] = e[k] * inv * rexp[k / 5] * cexp[k % 5];

    // Pass 2: weighted sum of the cached window vectors
    float4 olo = make_float4(0.f, 0.f, 0.f, 0.f);
    float4 ohi = make_float4(0.f, 0.f, 0.f, 0.f);
#pragma unroll
    for (int k = 0; k < KW_; ++k) {
        const float4* src = (const float4*)(myq + k * D_);
        const float4 lo = src[lane];
        const float4 hi = src[lane + 32];
        olo.x = fmaf(w[k], lo.x, olo.x); olo.y = fmaf(w[k], lo.y, olo.y);
        olo.z = fmaf(w[k], lo.z, olo.z); olo.w = fmaf(w[k], lo.w, olo.w);
        ohi.x = fmaf(w[k], hi.x, ohi.x); ohi.y = fmaf(w[k], hi.y, ohi.y);
        ohi.z = fmaf(w[k], hi.z, ohi.z); ohi.w = fmaf(w[k], hi.w, ohi.w);
    }

    float4* orow = (float4*)(out + (size_t)point * D_);
    orow[lane]      = olo;
    orow[lane + 32] = ohi;
}

// -----------------------------------------------------------------------------
extern "C" void kernel_launch(void* const* d_in, const int* in_sizes, int n_in,
                              void* d_out, int out_size, void* d_ws, size_t ws_size,
                              hipStream_t stream)
{
    const float* q   = (const float*)d_in[0];   // [B,H,W,D]
    const float* c_t = (const float*)d_in[1];   // [B,N,D]
    const float* p_t = (const float*)d_in[2];   // [B,N,2]
    const float* W_a = (const float*)d_in[3];   // [D,D]
    float* out = (float*)d_out;                 // [B,N,D]
    float* V   = (float*)d_ws;                  // [B*N, D] = 8 MB scratch

    // GEMM: (B*N=8192) x 256, one wave per 16x64 tile -> 2048 waves, 256 blocks
    gemm_f32_wmma<<<256, 256, 0, stream>>>(c_t, W_a, V);

    // Attention: one wave per point, 4 waves/block, 60 KB dynamic LDS
    local_attn<<<(B_ * N_) / 4, 128, 4 * KW_ * D_ * sizeof(float), stream>>>(q, p_t, V, out);
}